// GATLocalizationModel_17343077941393
// MI455X (gfx1250) — compile-verified
//
#include <hip/hip_runtime.h>
#include <cstdint>
#include <cstddef>

// ---------------------------------------------------------------------------
// GATv2 (2 layers) + MLP head for MI455X (gfx1250, wave32).
// Dense GEMMs use V_WMMA_F32_16X16X4_F32 (full fp32 matrix pipe).
// Edge attention: 1 wave per edge, float4-coalesced gathers (L2-resident
// feature tables), shfl head reductions, ordered-int atomicMax softmax.
// ---------------------------------------------------------------------------

typedef __attribute__((ext_vector_type(2))) float v2f;
typedef __attribute__((ext_vector_type(8))) float v8f;

#define NEG_SLOPE 0.2f

__device__ __forceinline__ unsigned int encOrd(float f) {
  unsigned int u = __float_as_uint(f);
  return (u & 0x80000000u) ? ~u : (u | 0x80000000u);
}
__device__ __forceinline__ float decOrd(unsigned int e) {
  unsigned int u = (e & 0x80000000u) ? (e & 0x7FFFFFFFu) : ~e;
  return __uint_as_float(u);
}

// ---------------------------------------------------------------------------
// C[M x Nout] = act(A[M x K] @ W[Nout x K]^T + bias)
// One wave computes a 16x16 tile via V_WMMA_F32_16X16X4_F32 over K/4 steps.
// blockDim.x = 32 * (Nout/16)  (Nout/16 <= 8). gridDim.x = ceil(M/16).
// A-matrix lane layout (ISA 7.12.2, 32-bit A 16x4): lanes 0-15 hold K=0,1;
// lanes 16-31 hold K=2,3 (2 VGPRs). B mirrors with N striped across lanes.
// D layout: VGPR r: lanes0-15 -> M=r, lanes16-31 -> M=r+8, N = lane&15.
// ---------------------------------------------------------------------------
__global__ void gemm_wmma(const float* __restrict__ A, const float* __restrict__ W,
                          const float* __restrict__ bias, float* __restrict__ C,
                          int M, int K, int Nout, int relu) {
  const int lane = threadIdx.x & 31;
  const int wave = threadIdx.x >> 5;
  const int row0 = blockIdx.x * 16;
  const int col0 = wave * 16;
  if (col0 >= Nout || row0 >= M) return;   // wave-uniform exit
  const int half = lane >> 4;              // 0: K=0,1   1: K=2,3
  const int l16  = lane & 15;
  int arow = row0 + l16; if (arow >= M) arow = M - 1;   // clamp (M%16==0 here)
  const float* __restrict__ Ar = A + (size_t)arow * K;
  const float* __restrict__ Wr = W + (size_t)(col0 + l16) * K;
  v8f acc = {0.f, 0.f, 0.f, 0.f, 0.f, 0.f, 0.f, 0.f};
  for (int k = 0; k < K; k += 4) {
    const int kk = k + half * 2;
    v2f a; a.x = Ar[kk]; a.y = Ar[kk + 1];
    v2f b; b.x = Wr[kk]; b.y = Wr[kk + 1];
    acc = __builtin_amdgcn_wmma_f32_16x16x4_f32(false, a, false, b,
                                                (short)0, acc, false, false);
  }
  const int ocol = col0 + l16;
  const float bv = bias[ocol];
#pragma unroll
  for (int r = 0; r < 8; ++r) {
    const int orow = row0 + r + half * 8;
    if (orow < M) {
      float v = acc[r] + bv;
      if (relu) v = v > 0.f ? v : 0.f;
      C[(size_t)orow * Nout + ocol] = v;
    }
  }
}

// ---------------------------------------------------------------------------
// Deterministic mean of edge_attr: 2-stage tree reduction.
// ---------------------------------------------------------------------------
__global__ void reduce_sum_partial(const float* __restrict__ x, float* __restrict__ partial, int n) {
  __shared__ float sm[256];
  float s = 0.f;
  for (int i = blockIdx.x * 256 + threadIdx.x; i < n; i += 256 * gridDim.x) s += x[i];
  sm[threadIdx.x] = s; __syncthreads();
  for (int off = 128; off > 0; off >>= 1) {
    if ((int)threadIdx.x < off) sm[threadIdx.x] += sm[threadIdx.x + off];
    __syncthreads();
  }
  if (threadIdx.x == 0) partial[blockIdx.x] = sm[0];
}
__global__ void reduce_sum_final(const float* __restrict__ partial, float* __restrict__ out,
                                 int nb, float scale) {
  __shared__ float sm[256];
  sm[threadIdx.x] = ((int)threadIdx.x < nb) ? partial[threadIdx.x] : 0.f;
  __syncthreads();
  for (int off = 128; off > 0; off >>= 1) {
    if ((int)threadIdx.x < off) sm[threadIdx.x] += sm[threadIdx.x + off];
    __syncthreads();
  }
  if (threadIdx.x == 0) out[0] = sm[0] * scale;
}

__global__ void fill_f32(float* __restrict__ p, float v, int n) {
  int i = blockIdx.x * blockDim.x + threadIdx.x;
  if (i < n) p[i] = v;
}
__global__ void fill_u32(unsigned int* __restrict__ p, unsigned int v, int n) {
  int i = blockIdx.x * blockDim.x + threadIdx.x;
  if (i < n) p[i] = v;
}

// ---------------------------------------------------------------------------
// Pass 1: per-edge attention logits + segment max (ordered-uint atomicMax).
// One wave per edge; lane l covers channels 4l..4l+3 (coalesced float4).
// Head h owns lanes [h*gs, (h+1)*gs), gs = 32/H; shfl_xor reduces per head.
// Edges e >= E are the appended self loops (src=dst=e-E, ea = mean).
// ---------------------------------------------------------------------------
__global__ void edge_logits(const long long* __restrict__ ei, const float* __restrict__ ea,
                            const float* __restrict__ mean_ea,
                            const float* __restrict__ xl, const float* __restrict__ xr,
                            const float* __restrict__ We, const float* __restrict__ att,
                            float* __restrict__ alpha, unsigned int* __restrict__ amax,
                            int E, int Ntot, int H) {
  const int e = (int)(((size_t)blockIdx.x * blockDim.x + threadIdx.x) >> 5);
  if (e >= Ntot) return;
  const int lane = threadIdx.x & 31;
  int src, dst; float eav;
  if (e < E) { src = (int)ei[e]; dst = (int)ei[(size_t)E + e]; eav = ea[e]; }
  else       { src = e - E; dst = src; eav = mean_ea[0]; }
  const int c0 = lane << 2;
  const float4 vl = *(const float4*)(xl + (size_t)src * 128 + c0);
  const float4 vr = *(const float4*)(xr + (size_t)dst * 128 + c0);
  const float4 w  = *(const float4*)(We + c0);
  const float4 at = *(const float4*)(att + c0);
  float m, s = 0.f;
  m = vl.x + vr.x + eav * w.x; m = m > 0.f ? m : NEG_SLOPE * m; s += at.x * m;
  m = vl.y + vr.y + eav * w.y; m = m > 0.f ? m : NEG_SLOPE * m; s += at.y * m;
  m = vl.z + vr.z + eav * w.z; m = m > 0.f ? m : NEG_SLOPE * m; s += at.z * m;
  m = vl.w + vr.w + eav * w.w; m = m > 0.f ? m : NEG_SLOPE * m; s += at.w * m;
  const int gs = 32 / H;                       // lanes per head (8 or 32)
  for (int off = gs >> 1; off > 0; off >>= 1)  // xor < gs stays inside head group
    s += __shfl_xor(s, off, 32);
  if ((lane & (gs - 1)) == 0) {
    const int h = lane / gs;
    alpha[(size_t)e * H + h] = s;
    atomicMax(amax + (size_t)dst * H + h, encOrd(s));
  }
}

// Pass 2: ex = exp(alpha - max); denom[dst,h] += ex. One thread per (e,h).
__global__ void edge_exp(const long long* __restrict__ ei, float* __restrict__ alpha,
                         const unsigned int* __restrict__ amax, float* __restrict__ denom,
                         int E, int Ntot, int H) {
  const long long idx = (long long)blockIdx.x * blockDim.x + threadIdx.x;
  if (idx >= (long long)Ntot * H) return;
  const int e = (int)(idx / H);
  const int h = (int)(idx - (long long)e * H);
  const int dst = (e < E) ? (int)ei[(size_t)E + e] : e - E;
  const float a  = alpha[idx];
  const float mx = decOrd(amax[(size_t)dst * H + h]);
  const float ex = expf(a - mx);
  alpha[idx] = ex;
  atomicAdd(denom + (size_t)dst * H + h, ex);
}

// Pass 3: acc[dst] += xl[src] * (ex / (denom + eps)). One wave per edge.
__global__ void edge_scatter(const long long* __restrict__ ei, const float* __restrict__ alpha,
                             const float* __restrict__ denom, const float* __restrict__ xl,
                             float* __restrict__ acc, int E, int Ntot, int H) {
  const int e = (int)(((size_t)blockIdx.x * blockDim.x + threadIdx.x) >> 5);
  if (e >= Ntot) return;
  const int lane = threadIdx.x & 31;
  int src, dst;
  if (e < E) { src = (int)ei[e]; dst = (int)ei[(size_t)E + e]; }
  else       { src = e - E; dst = src; }
  const int c0 = lane << 2;
  const int gs = 32 / H;
  const int h  = lane / gs;
  const float ex = alpha[(size_t)e * H + h];
  const float dn = denom[(size_t)dst * H + h];
  const float cf = ex / (dn + 1e-16f);
  const float4 vl = *(const float4*)(xl + (size_t)src * 128 + c0);
  float* o = acc + (size_t)dst * 128 + c0;
  atomicAdd(o + 0, vl.x * cf);
  atomicAdd(o + 1, vl.y * cf);
  atomicAdd(o + 2, vl.z * cf);
  atomicAdd(o + 3, vl.w * cf);
}

// acc = elu(acc + bias[c] + res), 128 channels, in place.
__global__ void epilogue_elu(float* __restrict__ acc, const float* __restrict__ bias,
                             const float* __restrict__ res, int total) {
  const int i = blockIdx.x * blockDim.x + threadIdx.x;
  if (i >= total) return;
  const float v = acc[i] + bias[i & 127] + res[i];
  acc[i] = v > 0.f ? v : (expf(v) - 1.f);
}

// Final tiny layer: out[N,3] = y2[N,32] @ fc3_W[3,32]^T + b.
__global__ void fc3_kernel(const float* __restrict__ y2, const float* __restrict__ W,
                           const float* __restrict__ b, float* __restrict__ out, int N) {
  const int i = blockIdx.x * blockDim.x + threadIdx.x;
  if (i >= N) return;
  const float* r = y2 + (size_t)i * 32;
#pragma unroll
  for (int j = 0; j < 3; ++j) {
    float s = b[j];
#pragma unroll
    for (int k = 0; k < 32; ++k) s += r[k] * W[j * 32 + k];
    out[i * 3 + j] = s;
  }
}

// ---------------------------------------------------------------------------
extern "C" void kernel_launch(void* const* d_in, const int* in_sizes, int n_in,
                              void* d_out, int out_size, void* d_ws, size_t ws_size,
                              hipStream_t stream) {
  const float*     x      = (const float*)d_in[0];
  const long long* ei     = (const long long*)d_in[1];   // int64 [2,E]
  const float*     ea     = (const float*)d_in[2];
  const float* g1_Wl  = (const float*)d_in[3];
  const float* g1_bl  = (const float*)d_in[4];
  const float* g1_Wr  = (const float*)d_in[5];
  const float* g1_br  = (const float*)d_in[6];
  const float* g1_We  = (const float*)d_in[7];
  const float* g1_att = (const float*)d_in[8];
  const float* g1_bias= (const float*)d_in[9];
  const float* g2_Wl  = (const float*)d_in[10];
  const float* g2_bl  = (const float*)d_in[11];
  const float* g2_Wr  = (const float*)d_in[12];
  const float* g2_br  = (const float*)d_in[13];
  const float* g2_We  = (const float*)d_in[14];
  const float* g2_att = (const float*)d_in[15];
  const float* g2_bias= (const float*)d_in[16];
  const float* res1_W = (const float*)d_in[17];
  const float* res1_b = (const float*)d_in[18];
  const float* fc1_W  = (const float*)d_in[19];
  const float* fc1_b  = (const float*)d_in[20];
  const float* fc2_W  = (const float*)d_in[21];
  const float* fc2_b  = (const float*)d_in[22];
  const float* fc3_W  = (const float*)d_in[23];
  const float* fc3_b  = (const float*)d_in[24];

  const int N    = in_sizes[0] / 64;   // 50000
  const int E    = in_sizes[2];        // 800000
  const int Ntot = E + N;              // with self loops

  // Workspace layout (floats); ~118 MB total.
  float* ws = (float*)d_ws;
  const size_t NS = (size_t)N * 128;
  float* b0 = ws;                 // xl (both layers) / y1
  float* b1 = b0 + NS;            // xr (both layers) / y2
  float* b2 = b1 + NS;            // layer1 acc -> h (in place)
  float* b3 = b2 + NS;            // layer1 res; layer2 acc -> h2 (in place)
  float* alpha = b3 + NS;                         // [Ntot * 4] max
  float* denom = alpha + (size_t)Ntot * 4;        // [N * 4]
  unsigned int* amax = (unsigned int*)(denom + (size_t)N * 4);  // [N * 4]
  float* partial = (float*)(amax + (size_t)N * 4);              // [256]
  float* mean_ea = partial + 256;                               // [1]

  const int totNS = (int)NS;
  const dim3 gRow((N + 15) / 16);
  const int ewBlocks = (Ntot + 7) / 8;   // 8 waves (edges) per 256-thr block

  // mean(edge_attr) — deterministic 2-stage reduction
  reduce_sum_partial<<<256, 256, 0, stream>>>(ea, partial, E);
  reduce_sum_final<<<1, 256, 0, stream>>>(partial, mean_ea, 256, 1.0f / (float)E);

  // ---------------- Layer 1 (H=4, C=32) ----------------
  gemm_wmma<<<gRow, 256, 0, stream>>>(x, g1_Wl, g1_bl, b0, N, 64, 128, 0);
  gemm_wmma<<<gRow, 256, 0, stream>>>(x, g1_Wr, g1_br, b1, N, 64, 128, 0);
  gemm_wmma<<<gRow, 256, 0, stream>>>(x, res1_W, res1_b, b3, N, 64, 128, 0);
  fill_f32<<<(totNS + 255) / 256, 256, 0, stream>>>(b2, 0.f, totNS);
  fill_f32<<<(N * 4 + 255) / 256, 256, 0, stream>>>(denom, 0.f, N * 4);
  fill_u32<<<(N * 4 + 255) / 256, 256, 0, stream>>>(amax, 0u, N * 4);
  edge_logits<<<ewBlocks, 256, 0, stream>>>(ei, ea, mean_ea, b0, b1, g1_We, g1_att,
                                            alpha, amax, E, Ntot, 4);
  {
    long long th = (long long)Ntot * 4;
    edge_exp<<<(unsigned)((th + 255) / 256), 256, 0, stream>>>(ei, alpha, amax, denom, E, Ntot, 4);
  }
  edge_scatter<<<ewBlocks, 256, 0, stream>>>(ei, alpha, denom, b0, b2, E, Ntot, 4);
  epilogue_elu<<<(totNS + 255) / 256, 256, 0, stream>>>(b2, g1_bias, b3, totNS);  // b2 = h

  // ---------------- Layer 2 (H=1, C=128) ----------------
  gemm_wmma<<<gRow, 256, 0, stream>>>(b2, g2_Wl, g2_bl, b0, N, 128, 128, 0);
  gemm_wmma<<<gRow, 256, 0, stream>>>(b2, g2_Wr, g2_br, b1, N, 128, 128, 0);
  fill_f32<<<(totNS + 255) / 256, 256, 0, stream>>>(b3, 0.f, totNS);
  fill_f32<<<(N + 255) / 256, 256, 0, stream>>>(denom, 0.f, N);
  fill_u32<<<(N + 255) / 256, 256, 0, stream>>>(amax, 0u, N);
  edge_logits<<<ewBlocks, 256, 0, stream>>>(ei, ea, mean_ea, b0, b1, g2_We, g2_att,
                                            alpha, amax, E, Ntot, 1);
  edge_exp<<<(Ntot + 255) / 256, 256, 0, stream>>>(ei, alpha, amax, denom, E, Ntot, 1);
  edge_scatter<<<ewBlocks, 256, 0, stream>>>(ei, alpha, denom, b0, b3, E, Ntot, 1);
  epilogue_elu<<<(totNS + 255) / 256, 256, 0, stream>>>(b3, g2_bias, b2, totNS);  // b3 = h2

  // ---------------- MLP head ----------------
  gemm_wmma<<<gRow, 64, 0, stream>>>(b3, fc1_W, fc1_b, b0, N, 128, 32, 1);  // y1
  gemm_wmma<<<gRow, 64, 0, stream>>>(b0, fc2_W, fc2_b, b1, N, 32, 32, 1);   // y2
  fc3_kernel<<<(N + 255) / 256, 256, 0, stream>>>(b1, fc3_W, fc3_b, (float*)d_out, N);
}